// GwcVolume_34437047779563
// MI455X (gfx1250) — compile-verified
//
#include <hip/hip_runtime.h>

typedef __attribute__((ext_vector_type(2))) float v2f;
typedef __attribute__((ext_vector_type(8))) float v8f;

// Problem constants (from reference setup)
#define BB    2
#define CC    320
#define GG    40
#define CPG   8
#define HH    96
#define WW    312
#define DISP  48
#define HWSZ  (HH * WW)

#define CHUNK   128           // output w columns per block
#define WPL     132           // lT row pitch (floats), 16B-aligned rows
#define RCOLS   176           // rT valid cols: w' in [w_start-48, w_start+128)
#define WPR     180           // rT row pitch (floats)
#define OPITCH  132           // outT row pitch (floats)
#define OROWS   79            // 15 trash + 48 valid + 16 trash rows (d offset +15)

__global__ __launch_bounds__(256) void gwc_volume_wmma(
    const float* __restrict__ L, const float* __restrict__ R, float* __restrict__ out)
{
    __shared__ float lT[CPG * WPL];        // lT[c][w_local], pre-scaled by 1/8
    __shared__ float rT[CPG * WPR];        // rT[c][w'_local], zero-guarded
    __shared__ float outT[OROWS * OPITCH]; // outT[15+d][w_local], trash-padded

    const int tid     = threadIdx.x;
    const int chunk   = blockIdx.x;        // 0..2
    const int h       = blockIdx.y;        // 0..95
    const int bg      = blockIdx.z;        // 0..79  (b*40+g)
    const int b       = bg / GG;
    const int g       = bg - b * GG;
    const int w_start = chunk * CHUNK;

    const float* lbase = L + ((size_t)(b * CC + g * CPG) * HH + h) * WW;
    const float* rbase = R + ((size_t)(b * CC + g * CPG) * HH + h) * WW;

    // ---- Stage L/R row slices into LDS ([c][w], float4 both sides) ----
    {   // lT: 8 rows x 32 quads = 256 quads -> exactly one per thread
        int c = tid >> 5, q = tid & 31;
        int w = w_start + q * 4;
        float4 v = make_float4(0.f, 0.f, 0.f, 0.f);
        if (w + 3 < WW) v = *(const float4*)&lbase[(size_t)c * HWSZ + w];
        v.x *= 0.125f; v.y *= 0.125f; v.z *= 0.125f; v.w *= 0.125f;  // fold mean
        *(float4*)&lT[c * WPL + q * 4] = v;
    }
    for (int e = tid; e < CPG * (RCOLS / 4); e += 256) {  // rT: 8 x 44 quads
        int c = e / (RCOLS / 4), q = e - c * (RCOLS / 4);
        int wp = w_start - DISP + q * 4;                  // w' of first lane elem
        float4 v = make_float4(0.f, 0.f, 0.f, 0.f);
        if (wp >= 0 && wp + 3 < WW) v = *(const float4*)&rbase[(size_t)c * HWSZ + wp];
        *(float4*)&rT[c * WPR + q * 4] = v;               // zeros reproduce ref pad
    }
    __syncthreads();

    // ---- WMMA band tiles: D[m,n] = (1/8) sum_c l[c,w0+m] * r[c,w0-s+n] ----
    const int wave = tid >> 5;
    const int lane = tid & 31;
    const int s    = (wave & 3) * 16;      // disparity base of this wave's tiles
    const int half = lane >> 4;            // lanes 16..31: K+2 / M+8 halves
    const int mn   = lane & 15;            // A: row m ; B: col n
    const int kc   = half * 2;             // K row base within cpg

    for (int t = (wave >> 2); t < CHUNK / 16; t += 2) {
        const int w0l = t * 16;
        const float* aP = &lT[kc * WPL + w0l + mn];
        const float* bP = &rT[kc * WPR + w0l + DISP - s + mn];

        v2f a0 = { aP[0],       aP[WPL]     };   // K = kc, kc+1
        v2f b0 = { bP[0],       bP[WPR]     };
        v2f a1 = { aP[4 * WPL], aP[5 * WPL] };   // K = kc+4, kc+5
        v2f b1 = { bP[4 * WPR], bP[5 * WPR] };

        v8f acc = {};
        acc = __builtin_amdgcn_wmma_f32_16x16x4_f32(false, a0, false, b0,
                                                    (short)0, acc, false, false);
        acc = __builtin_amdgcn_wmma_f32_16x16x4_f32(false, a1, false, b1,
                                                    (short)0, acc, false, false);

        // D[m,n] -> (d = s+m-n, w = w0+m). Rows offset +15 so every write is
        // in-bounds (trash rows absorb d<0 / d>=48; each slot written once).
        // row and col both advance with v -> single base + imm offsets.
        const int row0 = 15 + s + half * 8 - mn;
        const int col0 = w0l + half * 8;
        float* oP = &outT[row0 * OPITCH + col0];
        #pragma unroll
        for (int v = 0; v < 8; ++v)
            oP[v * (OPITCH + 1)] = acc[v];
    }
    __syncthreads();

    // ---- Coalesced float4 stores of the [48][CHUNK] staging tile ----
    const size_t obase = (size_t)bg * DISP * HWSZ + (size_t)h * WW;
    for (int idx = tid; idx < DISP * (CHUNK / 4); idx += 256) {
        int d  = idx >> 5;                  // CHUNK/4 == 32
        int w4 = idx & 31;
        int w  = w_start + w4 * 4;
        if (w >= WW) continue;              // chunk 2 tail (quad-aligned)
        float4 v = *(const float4*)&outT[(15 + d) * OPITCH + w4 * 4];
        *(float4*)(out + obase + (size_t)d * HWSZ + w) = v;
    }
}

extern "C" void kernel_launch(void* const* d_in, const int* in_sizes, int n_in,
                              void* d_out, int out_size, void* d_ws, size_t ws_size,
                              hipStream_t stream) {
    (void)in_sizes; (void)n_in; (void)d_ws; (void)ws_size; (void)out_size;
    const float* L = (const float*)d_in[0];
    const float* R = (const float*)d_in[1];
    float* out = (float*)d_out;
    dim3 grid(3, HH, BB * GG);   // (w-chunk, h, b*G+g) = (3, 96, 80)
    gwc_volume_wmma<<<grid, 256, 0, stream>>>(L, R, out);
}